// STRNNCell_70712341561407
// MI455X (gfx1250) — compile-verified
//
#include <hip/hip_runtime.h>
#include <hip/hip_bf16.h>

// STRNN cell for MI455X (gfx1250), fp32 WMMA path.
//
// Shapes: B=256, C=50, INPUT=128, INTER=64, HIDDEN=128, SLOTS=11.
// Roofline: ~0.43 GFLOP vs ~7 MB moved -> memory-floor ~0.3us @ 23.3 TB/s.
// Full f32 precision via V_WMMA_F32_16X16X4_F32; the ~8x FLOP inflation from
// the per-slot replay (11 weight slots, row-select after each GEMM) is free.
// This revision software-pipelines the B-operand loads (depth 2) so the
// per-k-step `s_wait_loadcnt 0x0` stall observed in round 1 is covered by
// in-flight WMMAs (occupancy is ~1 wave/SIMD here, so intra-wave ILP is the
// only latency-hiding mechanism available).

typedef __attribute__((ext_vector_type(2))) float v2f;
typedef __attribute__((ext_vector_type(8))) float v8f;

static __device__ __forceinline__ v8f wmma_f32(v2f a, v2f b, v8f c) {
  // 8 args: (neg_a, A, neg_b, B, c_mod, C, reuse_a, reuse_b)
  return __builtin_amdgcn_wmma_f32_16x16x4_f32(false, a, false, b, (short)0, c,
                                               false, false);
}

#define NB 256
#define NC 50
#define NIN 128
#define NINTER 64
#define NHID 128
#define NSLOT 11

// Load the B operands (v2f per n-tile) for one k-step of a 16x16x4 WMMA.
// Lane layout assumption (mirrors documented A/C layouts): lane&15 = N,
// lane>=16 selects K+2; VGPR pair = consecutive K rows.
template <int NT, int STRIDE>
static __device__ __forceinline__ void load_bt(const float* __restrict__ wb,
                                               int krow, int l15, v2f* bvv) {
  const float* bp = wb + (size_t)krow * STRIDE + l15;
#pragma unroll
  for (int nt = 0; nt < NT; ++nt) {
    bvv[nt].x = bp[nt * 16];            // B[K=krow  ][N=nt*16+l15]
    bvv[nt].y = bp[nt * 16 + STRIDE];   // B[K=krow+1][N=nt*16+l15]
  }
}

// Kernel 1: per-batch masked context transform.
// grid = B workgroups; block = 128 threads = 4 waves; wave w owns contexts
// [16w, 16w+16) of its batch (padded past C=50 via slot=255 sentinel).
__global__ __launch_bounds__(128) void strnn_ctx_kernel(
    const float* __restrict__ x,            // (B, C, INPUT)
    const int* __restrict__ tctx,           // (B, C)
    const int* __restrict__ dctx,           // (B, C)
    const unsigned char* __restrict__ cmask,// (B, C) bool (1 byte)
    const float* __restrict__ tw,           // (11, INPUT, INTER)
    const float* __restrict__ dw,           // (11, INTER, HIDDEN)
    float* __restrict__ xc_sum)             // (B, HIDDEN) out
{
  __shared__ float t_buf[4][16 * NINTER];   // per-wave 16x64 t tile
  __shared__ float partial[4][NHID];        // per-wave masked row-sums

  const int b    = blockIdx.x;
  const int wave = threadIdx.x >> 5;
  const int lane = threadIdx.x & 31;
  const int l15  = lane & 15;
  const int h16  = lane >> 4;
  const int c0   = wave * 16;

  // Effective slot per D-layout element e (row = e + 8*h16); 255 = padded or
  // masked-out row (never matches a slot -> contributes exactly zero).
  int eff_t[8], eff_d[8];
  unsigned bits_t = 0u, bits_d = 0u;
#pragma unroll
  for (int e = 0; e < 8; ++e) {
    const int row = e + 8 * h16;
    const int c   = c0 + row;
    int ts = 255, ds = 255;
    if (c < NC) {
      const int idx = b * NC + c;
      if (cmask[idx] != 0) { ts = tctx[idx]; ds = dctx[idx]; }
    }
    eff_t[e] = ts;
    eff_d[e] = ds;
    if (ts < 32) bits_t |= 1u << ts;
    if (ds < 32) bits_d |= 1u << ds;
  }
  // lanes 0-15 carry rows 0-7, lanes 16-31 rows 8-15: one xor-16 makes the
  // present-slot masks wave-uniform (uniform branch -> EXEC stays all-1s).
  bits_t |= __shfl_xor(bits_t, 16, 32);
  bits_d |= __shfl_xor(bits_d, 16, 32);

  // ---- Stage 1 A tile in registers: A[M=l15][K = 4kk + 2*h16 + {0,1}] ----
  v2f a_reg[32];
  {
    const int  c = c0 + l15;
    const bool v = (c < NC);
    const float* xr = x + ((size_t)b * NC + (v ? c : 0)) * NIN;
    const v2f z2 = {0.f, 0.f};
#pragma unroll
    for (int kk = 0; kk < 32; ++kk)
      a_reg[kk] = v ? *(const v2f*)(xr + 4 * kk + 2 * h16) : z2;
  }

  // ---- Stage 1: t = x @ TW[slot], row-selected per present slot ----
  const v8f z8 = {0.f, 0.f, 0.f, 0.f, 0.f, 0.f, 0.f, 0.f};
  v8f acc1[4];
#pragma unroll
  for (int nt = 0; nt < 4; ++nt) acc1[nt] = z8;

  for (int s = 0; s < NSLOT; ++s) {
    if (!((bits_t >> s) & 1u)) continue;      // wave-uniform skip
    v8f d[4];
#pragma unroll
    for (int nt = 0; nt < 4; ++nt) d[nt] = z8;
    const float* wb = tw + (size_t)s * NIN * NINTER;

    // Depth-2 software pipeline: B for k-steps kk+1, kk+2 in flight while
    // the 4 WMMAs of k-step kk execute.
    v2f bv[3][4];
    load_bt<4, NINTER>(wb, 0 + 2 * h16, l15, bv[0]);
    load_bt<4, NINTER>(wb, 4 + 2 * h16, l15, bv[1]);
#pragma unroll
    for (int kk = 0; kk < 32; ++kk) {
      if (kk + 2 < 32)
        load_bt<4, NINTER>(wb, 4 * (kk + 2) + 2 * h16, l15, bv[(kk + 2) % 3]);
#pragma unroll
      for (int nt = 0; nt < 4; ++nt)
        d[nt] = wmma_f32(a_reg[kk], bv[kk % 3][nt], d[nt]);
    }
#pragma unroll
    for (int nt = 0; nt < 4; ++nt)
#pragma unroll
      for (int e = 0; e < 8; ++e)
        acc1[nt][e] += (eff_t[e] == s) ? d[nt][e] : 0.0f;
  }

  // ---- D-layout -> A-layout via LDS (same-wave roundtrip) ----
  float* tb = t_buf[wave];
#pragma unroll
  for (int nt = 0; nt < 4; ++nt)
#pragma unroll
    for (int e = 0; e < 8; ++e)
      tb[(e + 8 * h16) * NINTER + nt * 16 + l15] = acc1[nt][e];

  v2f a2[16];
#pragma unroll
  for (int kk = 0; kk < 16; ++kk)
    a2[kk] = *(const v2f*)(tb + l15 * NINTER + 4 * kk + 2 * h16);

  // ---- Stage 2: xc = t @ DW[slot], row-selected per present slot ----
  v8f acc2[8];
#pragma unroll
  for (int nt = 0; nt < 8; ++nt) acc2[nt] = z8;

  for (int s = 0; s < NSLOT; ++s) {
    if (!((bits_d >> s) & 1u)) continue;
    v8f d[8];
#pragma unroll
    for (int nt = 0; nt < 8; ++nt) d[nt] = z8;
    const float* wb = dw + (size_t)s * NINTER * NHID;

    // Depth-1 pipeline here (acc2[8]+d[8]+a2 already ~160 VGPRs): B for
    // k-step kk+1 in flight behind 8 WMMAs of k-step kk.
    v2f bv[2][8];
    load_bt<8, NHID>(wb, 2 * h16, l15, bv[0]);
#pragma unroll
    for (int kk = 0; kk < 16; ++kk) {
      if (kk + 1 < 16)
        load_bt<8, NHID>(wb, 4 * (kk + 1) + 2 * h16, l15, bv[(kk + 1) & 1]);
#pragma unroll
      for (int nt = 0; nt < 8; ++nt)
        d[nt] = wmma_f32(a2[kk], bv[kk & 1][nt], d[nt]);
    }
#pragma unroll
    for (int nt = 0; nt < 8; ++nt)
#pragma unroll
      for (int e = 0; e < 8; ++e)
        acc2[nt][e] += (eff_d[e] == s) ? d[nt][e] : 0.0f;
  }

  // ---- Masked sum over the 16 rows, then across the 4 waves (fixed order,
  // bit-deterministic; masked rows are already zero) ----
#pragma unroll
  for (int nt = 0; nt < 8; ++nt) {
    float v = 0.f;
#pragma unroll
    for (int e = 0; e < 8; ++e) v += acc2[nt][e];   // rows within lane-half
    v += __shfl_xor(v, 16, 32);                     // combine the two halves
    if (h16 == 0) partial[wave][nt * 16 + l15] = v;
  }
  __syncthreads();

  if (threadIdx.x < NHID) {
    const int col = threadIdx.x;
    xc_sum[(size_t)b * NHID + col] =
        partial[0][col] + partial[1][col] + partial[2][col] + partial[3][col];
  }
}

// Kernel 2: out = sigmoid(xc_sum + h @ HW), WMMA 16x16 tiles, K=128.
// grid = (B/16, HIDDEN/16), block = 1 wave.
__global__ __launch_bounds__(32) void strnn_out_kernel(
    const float* __restrict__ h,       // (B, HIDDEN)
    const float* __restrict__ hw,      // (HIDDEN, HIDDEN)
    const float* __restrict__ xc_sum,  // (B, HIDDEN)
    float* __restrict__ out)           // (B, HIDDEN)
{
  const int lane = threadIdx.x & 31;
  const int l15  = lane & 15;
  const int h16  = lane >> 4;
  const int b0   = blockIdx.x * 16;
  const int n0   = blockIdx.y * 16;

  const float* ha = h + (size_t)(b0 + l15) * NHID + 2 * h16;
  const float* hb = hw + n0 + l15;

  // Depth-2 pipeline on both operands of the serial accumulation chain.
  v2f av[3], bvv[3];
#pragma unroll
  for (int p = 0; p < 2; ++p) {
    const int k = 4 * p + 2 * h16;
    av[p]    = *(const v2f*)(ha + 4 * p);
    bvv[p].x = hb[(size_t)k * NHID];
    bvv[p].y = hb[(size_t)(k + 1) * NHID];
  }

  v8f d = {0.f, 0.f, 0.f, 0.f, 0.f, 0.f, 0.f, 0.f};
#pragma unroll
  for (int kk = 0; kk < 32; ++kk) {
    if (kk + 2 < 32) {
      const int kn = 4 * (kk + 2) + 2 * h16;
      av[(kk + 2) % 3]    = *(const v2f*)(ha + 4 * (kk + 2));
      bvv[(kk + 2) % 3].x = hb[(size_t)kn * NHID];
      bvv[(kk + 2) % 3].y = hb[(size_t)(kn + 1) * NHID];
    }
    d = wmma_f32(av[kk % 3], bvv[kk % 3], d);
  }

#pragma unroll
  for (int e = 0; e < 8; ++e) {
    const int bb = b0 + e + 8 * h16;
    const int n  = n0 + l15;
    const float v = d[e] + xc_sum[(size_t)bb * NHID + n];
    out[(size_t)bb * NHID + n] = 1.0f / (1.0f + __expf(-v));
  }
}

extern "C" void kernel_launch(void* const* d_in, const int* in_sizes, int n_in,
                              void* d_out, int out_size, void* d_ws, size_t ws_size,
                              hipStream_t stream) {
  const float*         x     = (const float*)d_in[0];         // x_context
  const int*           tctx  = (const int*)d_in[1];           // time_context
  const int*           dctx  = (const int*)d_in[2];           // dist_context
  const unsigned char* cmask = (const unsigned char*)d_in[3]; // context_mask (bool)
  const float*         h     = (const float*)d_in[4];         // h
  const float*         tw    = (const float*)d_in[5];         // time_weights
  const float*         dw    = (const float*)d_in[6];         // dist_weights
  const float*         hw    = (const float*)d_in[7];         // hidden_weights
  float*               out   = (float*)d_out;
  float*               xcsum = (float*)d_ws;                  // (B, HIDDEN) scratch

  // Kernel 1 fully overwrites xc_sum (plain stores, no accumulation into
  // poisoned scratch), kernel 2 consumes it; stream order serializes them.
  strnn_ctx_kernel<<<NB, 128, 0, stream>>>(x, tctx, dctx, cmask, tw, dw, xcsum);
  dim3 g2(NB / 16, NHID / 16);
  strnn_out_kernel<<<g2, 32, 0, stream>>>(h, hw, xcsum, out);
}